// VectorizedDOOMRaycaster_32512902431226
// MI455X (gfx1250) — compile-verified
//
#include <hip/hip_runtime.h>
#include <stdint.h>

// ---- problem constants (match reference) ----
#define FB_WIDTH   80
#define MAP_SIZE   16
#define BATCH      4096
#define STEP_SIZE  0.1f
#define MAX_STEPS  200
#define MAX_DIST   20.0f

typedef __attribute__((ext_vector_type(2))) float v2f;
typedef __attribute__((ext_vector_type(8))) float v8f;

// One wave (32 lanes) per 16(player) x 16(ray) tile.
//  - angles for the whole tile computed with one V_WMMA_F32_16X16X4_F32
//    (exact: products are pa*1 and 1*ro, zeros elsewhere)
//  - WMMA C/D layout: vgpr j, lanes 0-15 -> (M=j,   N=lane)
//                             lanes 16-31 -> (M=j+8, N=lane-16)
//    so each lane marches 8 rays (ILP=8).
//  - 16x16 map (1KB) pulled into LDS with gfx1250 async-to-LDS b128 ops,
//    overlapped with the WMMA / sincos setup, fenced by s_wait_asynccnt.
__global__ __launch_bounds__(32) void doom_raycast_kernel(
    const float* __restrict__ player_x,
    const float* __restrict__ player_y,
    const float* __restrict__ player_angle,
    const float* __restrict__ ray_offsets,
    const float* __restrict__ game_map,
    float* __restrict__ out)
{
    __shared__ float smap[MAP_SIZE * MAP_SIZE];

    const int lane  = threadIdx.x;            // 0..31, wave32
    const int tile  = blockIdx.x;             // (4096/16)*(80/16) = 1280 tiles
    const int bTile = (tile / (FB_WIDTH / 16)) * 16;
    const int wTile = (tile % (FB_WIDTH / 16)) * 16;
    const int n     = lane & 15;              // ray column within tile
    const int half  = lane >> 4;              // 0: M=0..7, 1: M=8..15

    // ---- async copy of the map into LDS (ASYNCcnt path) --------------------
    // per lane: 16B at lane*16, plus 16B at lane*16+512 -> 2*512B = 1KB total.
    // INST_OFFSET is added to both the LDS and the global address (ISA 08 §4.4),
    // so the second instruction reuses the same base VGPRs.
    {
        unsigned           ldsOff = (unsigned)(uintptr_t)(void*)smap + (unsigned)lane * 16u;
        unsigned long long gAddr  = (unsigned long long)(uintptr_t)game_map
                                    + (unsigned long long)lane * 16ull;
        asm volatile("global_load_async_to_lds_b128 %0, %1, off"
                     :: "v"(ldsOff), "v"(gAddr) : "memory");
        asm volatile("global_load_async_to_lds_b128 %0, %1, off offset:512"
                     :: "v"(ldsOff), "v"(gAddr) : "memory");
    }

    // ---- angles for the tile: outer sum pa[m] + ro[n] ----------------------
    v8f ang;
#if __has_builtin(__builtin_amdgcn_wmma_f32_16x16x4_f32)
    {
        // A (16x4): A[m,0]=pa[m], A[m,1]=1, A[m,2..3]=0
        //   layout: lanes 0-15 hold K=0 (v0) / K=1 (v1); lanes 16-31 hold K=2/K=3
        // B (4x16): B[0,n]=1, B[1,n]=ro[n], B[2..3,n]=0
        //   layout mirrors A: lanes 0-15 hold K=0/K=1; lanes 16-31 hold K=2/K=3
        float pa = player_angle[bTile + n];   // row m = lane (lanes 16-31 unused)
        float ro = ray_offsets[wTile + n];    // col n = lane
        v2f A, Bm;
        A.x  = half ? 0.0f : pa;
        A.y  = half ? 0.0f : 1.0f;
        Bm.x = half ? 0.0f : 1.0f;
        Bm.y = half ? 0.0f : ro;
        v8f Cz = {};
        // D[m,n] = pa[m]*1 + 1*ro[n] + 0 + 0 + C  (exact in fp32)
        ang = __builtin_amdgcn_wmma_f32_16x16x4_f32(
                  false, A, false, Bm, (short)0, Cz, false, false);
    }
#else
    {
        float ro = ray_offsets[wTile + n];
#pragma unroll
        for (int j = 0; j < 8; ++j)
            ang[j] = player_angle[bTile + j + 8 * half] + ro;
    }
#endif

    // ---- per-ray state (8 rays per lane) -----------------------------------
    float px[8], py[8], dx[8], dy[8], rx[8], ry[8], dist[8];
#pragma unroll
    for (int j = 0; j < 8; ++j) {
        const int b = bTile + j + 8 * half;
        px[j]   = player_x[b];
        py[j]   = player_y[b];
        dx[j]   = __cosf(ang[j]);
        dy[j]   = __sinf(ang[j]);
        rx[j]   = px[j];
        ry[j]   = py[j];
        dist[j] = MAX_DIST;
    }
    unsigned hitmask = 0u;   // bit j: ray j has hit a wall (dist frozen)

    // map must be resident in LDS before the march
#if __has_builtin(__builtin_amdgcn_s_wait_asynccnt)
    __builtin_amdgcn_s_wait_asynccnt(0);
#else
    asm volatile("s_wait_asynccnt 0" ::: "memory");
#endif
    // single wave per workgroup -> no barrier needed after the wait

    // ---- march -------------------------------------------------------------
    for (int s = 0; s < MAX_STEPS; ++s) {
#pragma unroll
        for (int j = 0; j < 8; ++j) {
            rx[j] += dx[j] * STEP_SIZE;
            ry[j] += dy[j] * STEP_SIZE;
            int mx = (int)rx[j];             // trunc toward zero == astype(int32)
            int my = (int)ry[j];
            mx = mx < 0 ? 0 : (mx > MAP_SIZE - 1 ? MAP_SIZE - 1 : mx);
            my = my < 0 ? 0 : (my > MAP_SIZE - 1 ? MAP_SIZE - 1 : my);
            const bool wall = smap[my * MAP_SIZE + mx] > 0.5f;
            const float ddx = rx[j] - px[j];
            const float ddy = ry[j] - py[j];
#if __has_builtin(__builtin_amdgcn_sqrtf)
            const float nd = __builtin_amdgcn_sqrtf(ddx * ddx + ddy * ddy);
#else
            const float nd = sqrtf(ddx * ddx + ddy * ddy);
#endif
            const bool newhit = wall && !((hitmask >> j) & 1u);
            dist[j] = newhit ? nd : dist[j];
            hitmask |= wall ? (1u << j) : 0u;
        }
        // dist is frozen once hit: safe uniform early-exit when the whole
        // wave is done (wave32 ballot).
#if __has_builtin(__builtin_amdgcn_ballot_w32)
        if (__builtin_amdgcn_ballot_w32(hitmask != 0xFFu) == 0u) break;
#endif
    }

    // ---- store (lanes of one vgpr row are contiguous in memory) ------------
#pragma unroll
    for (int j = 0; j < 8; ++j) {
        const int b = bTile + j + 8 * half;
        out[b * FB_WIDTH + wTile + n] = dist[j];
    }
}

extern "C" void kernel_launch(void* const* d_in, const int* in_sizes, int n_in,
                              void* d_out, int out_size, void* d_ws, size_t ws_size,
                              hipStream_t stream) {
    (void)in_sizes; (void)n_in; (void)out_size; (void)d_ws; (void)ws_size;
    const float* player_x     = (const float*)d_in[0];   // (4096,)
    const float* player_y     = (const float*)d_in[1];   // (4096,)
    const float* player_angle = (const float*)d_in[2];   // (4096,)
    const float* ray_offsets  = (const float*)d_in[3];   // (80,)
    const float* game_map     = (const float*)d_in[4];   // (16,16)
    float*       out          = (float*)d_out;           // (4096,80)

    const int tiles = (BATCH / 16) * (FB_WIDTH / 16);    // 1280
    doom_raycast_kernel<<<dim3(tiles), dim3(32), 0, stream>>>(
        player_x, player_y, player_angle, ray_offsets, game_map, out);
}